// FreqPerturbation_52956946759736
// MI455X (gfx1250) — compile-verified
//
#include <hip/hip_runtime.h>

typedef __attribute__((ext_vector_type(2))) float v2f;
typedef __attribute__((ext_vector_type(8))) float v8f;

#define GAMMA_C 0.05f
#define EPS_C   1e-6f
#define SC512   0.04419417382415922f   /* 1/sqrt(512): ortho scale per 1-D pass */
#define PI2     6.283185307179586476925f

// ---------------------------------------------------------------------------
// One chained f32 WMMA: D = A(16x4) * B(4x16) + C  (wave32, CDNA5)
// ---------------------------------------------------------------------------
__device__ __forceinline__ v8f wmma4(v2f a, v2f b, v8f c) {
  return __builtin_amdgcn_wmma_f32_16x16x4_f32(
      /*neg_a=*/false, a, /*neg_b=*/false, b,
      /*c_mod=*/(short)0, c, /*reuse_a=*/false, /*reuse_b=*/false);
}

// Fast trig: all angles are pre-reduced to [0, 2pi) via integer mod, so the
// hardware v_sin_f32/v_cos_f32 path is accurate enough.
__device__ __forceinline__ void fsincos(float x, float* s, float* c) {
  __sincosf(x, s, c);
}

// ---------------------------------------------------------------------------
// 512-point complex FFT executed by ONE wave32 on its private LDS buffer.
// Four-step factorization 512 = 16 x 32:
//   Step1: Y[k1][n2] = sum_n1 D16[k1][n1] * X[n1][n2]      (WMMA, K=16)
//          then Y *= W512^(k1*n2)
//   Step2: Z[k1][k2] = sum_n2 Y[k1][n2] * W32[n2][k2]      (WMMA, K=32)
//          out[k1 + 16*k2] = Z[k1][k2] * scale
// DIR = -1 forward (e^{-i}), +1 inverse (e^{+i}).
// Fragment layouts per CDNA5 ISA 7.12.2:
//   A 16x4 : lane<16 -> (M=lane, K=4c+{0,1}),  lane>=16 -> (M=lane-16, K=4c+{2,3})
//   B 4x16 : lane<16 -> (K=4c+{0,1}, N=lane),  lane>=16 -> (K=4c+{2,3}, N=lane-16)
//   C/D    : VGPR j -> M = j (lanes 0-15) / j+8 (lanes 16-31), N = lane&15
// f32 WMMA has no A/B negation (NEG=CNeg only) -> subtraction via pre-negated
// imaginary data fragments.
// FORCEINLINE so the buf pointer provably derives from the kernel's __shared__
// array: accesses lower to ds_load/ds_store instead of generic FLAT.
// ---------------------------------------------------------------------------
template<int DIR>
__device__ __forceinline__ void fft512_wave(float2* buf, float scale) {
  const int lane = threadIdx.x & 31;
  const int nlo  = lane & 15;
  const int bse  = (lane < 16) ? 0 : 2;
  const float ds = (float)DIR;

  // ---- Step 1 constants: D16[m][k] = e^{DIR*2pi*i*m*k/16}, M=k1, K=n1
  v2f a_dr[4], a_di[4];
#pragma unroll
  for (int c = 0; c < 4; ++c) {
    int k0 = 4*c + bse;
    float t0 = PI2 * (float)((nlo *  k0   ) & 15) * (1.0f/16.0f);
    float t1 = PI2 * (float)((nlo * (k0+1)) & 15) * (1.0f/16.0f);
    float s0, c0, s1, c1;
    fsincos(t0, &s0, &c0);
    fsincos(t1, &s1, &c1);
    v2f dr; dr[0] = c0;    dr[1] = c1;
    v2f di; di[0] = ds*s0; di[1] = ds*s1;
    a_dr[c] = dr; a_di[c] = di;
  }

  // ---- Step 1: two 16-column tiles of X[n1][n2] = buf[n1*32+n2]
#pragma unroll
  for (int t = 0; t < 2; ++t) {
    const int col = 16*t + nlo;
    v2f bxr[4], bxi[4], bxin[4];
#pragma unroll
    for (int c = 0; c < 4; ++c) {
      float2 x0 = buf[(4*c + bse    )*32 + col];
      float2 x1 = buf[(4*c + bse + 1)*32 + col];
      v2f r;  r[0]  =  x0.x; r[1]  =  x1.x;
      v2f i;  i[0]  =  x0.y; i[1]  =  x1.y;
      v2f ni; ni[0] = -x0.y; ni[1] = -x1.y;
      bxr[c] = r; bxi[c] = i; bxin[c] = ni;
    }
    v8f cr = {}; v8f ci = {};
#pragma unroll
    for (int c = 0; c < 4; ++c) {
      cr = wmma4(a_dr[c], bxr[c],  cr);   // Re += Dr*Xr
      cr = wmma4(a_di[c], bxin[c], cr);   // Re += Di*(-Xi)
      ci = wmma4(a_dr[c], bxi[c],  ci);   // Im += Dr*Xi
      ci = wmma4(a_di[c], bxr[c],  ci);   // Im += Di*Xr
    }
    // twiddle W512^(m*col) and store Y back at buf[m*32+col]
    // (tile0 stores cols 0..15, tile1 reads cols 16..31 -> disjoint, wave-order safe)
#pragma unroll
    for (int j = 0; j < 8; ++j) {
      int m = j + ((lane < 16) ? 0 : 8);
      float th = PI2 * (float)(m * col) * (1.0f/512.0f);
      float sn, cs; fsincos(th, &sn, &cs);
      sn *= ds;
      float yr = cr[j], yi = ci[j];
      buf[m*32 + col] = make_float2(yr*cs - yi*sn, yr*sn + yi*cs);
    }
  }

  // ---- Step 2: load ALL of Y (A fragments, M=k1, K=n2) before any store
  v2f ayr[8], ayi[8], ayin[8];
#pragma unroll
  for (int c = 0; c < 8; ++c) {
    float2 y0 = buf[nlo*32 + 4*c + bse];
    float2 y1 = buf[nlo*32 + 4*c + bse + 1];
    v2f r;  r[0]  =  y0.x; r[1]  =  y1.x;
    v2f i;  i[0]  =  y0.y; i[1]  =  y1.y;
    v2f ni; ni[0] = -y0.y; ni[1] = -y1.y;
    ayr[c] = r; ayi[c] = i; ayin[c] = ni;
  }
#pragma unroll
  for (int t = 0; t < 2; ++t) {
    const int n = 16*t + nlo;            // k2
    v8f zr = {}; v8f zi = {};
#pragma unroll
    for (int c = 0; c < 8; ++c) {
      int k0 = 4*c + bse;                // n2
      float t0 = PI2 * (float)(( k0   *n) & 31) * (1.0f/32.0f);
      float t1 = PI2 * (float)(((k0+1)*n) & 31) * (1.0f/32.0f);
      float s0, c0, s1, c1;
      fsincos(t0, &s0, &c0);
      fsincos(t1, &s1, &c1);
      v2f bwr; bwr[0] = c0;    bwr[1] = c1;
      v2f bwi; bwi[0] = ds*s0; bwi[1] = ds*s1;
      zr = wmma4(ayr[c],  bwr, zr);
      zr = wmma4(ayin[c], bwi, zr);
      zi = wmma4(ayr[c],  bwi, zi);
      zi = wmma4(ayi[c],  bwr, zi);
    }
#pragma unroll
    for (int j = 0; j < 8; ++j) {
      int m = j + ((lane < 16) ? 0 : 8); // k1
      buf[n*16 + m] = make_float2(zr[j]*scale, zi[j]*scale);
    }
  }
}

#define WPB 4       // wave32s per block for the row kernels
#define CPB 16      // columns per block for the column kernels (16 waves)
#define CPITCH 517  // padded column pitch (float2) to avoid LDS bank conflicts

// ---- forward FFT of image rows (real -> complex), B128 coalesced -----------
__global__ __launch_bounds__(128) void k_fft_rows_fwd(const float* __restrict__ img,
                                                      float2* __restrict__ F) {
  __shared__ float2 lds[WPB * 512];
  const int wave = threadIdx.x >> 5, lane = threadIdx.x & 31;
  const size_t row = (size_t)blockIdx.x * WPB + wave;   // bc*512 + u
  float2* buf = lds + wave * 512;
  const float4* src4 = (const float4*)(img + row * 512);
#pragma unroll
  for (int j = 0; j < 4; ++j) {                 // 128 float4 per row
    float4 x = src4[lane + 32*j];
    int e = 4*(lane + 32*j);
    buf[e+0] = make_float2(x.x, 0.0f);
    buf[e+1] = make_float2(x.y, 0.0f);
    buf[e+2] = make_float2(x.z, 0.0f);
    buf[e+3] = make_float2(x.w, 0.0f);
  }
  fft512_wave<-1>(buf, SC512);
  float4* dst4 = (float4*)(F + row * 512);      // 2 complex per float4
#pragma unroll
  for (int j = 0; j < 8; ++j) {
    float2 a = buf[2*(lane + 32*j)], b = buf[2*(lane + 32*j) + 1];
    dst4[lane + 32*j] = make_float4(a.x, a.y, b.x, b.y);
  }
}

// ---- in-place FFT along columns, tiled for coalesced HBM access ------------
// Block = 512 threads (16 waves) = one 512-row x 16-column tile.
// Each lane moves a column-PAIR element as a float4 (B128); each wave FFTs one
// column held contiguously in padded LDS.
template<int DIR>
__global__ __launch_bounds__(512) void k_fft_cols(float2* __restrict__ F) {
  __shared__ float2 lds[CPB * CPITCH];   // ~66 KB of the 320 KB WGP LDS
  const int tid  = threadIdx.x;
  const int wave = tid >> 5;
  const int bc   = blockIdx.x >> 5;            // 32 tiles of 16 cols per image
  const int v0   = (blockIdx.x & 31) * CPB;
  float4* base4 = (float4*)(F + (size_t)bc * 262144 + v0);  // 8 float4 per row-tile
#pragma unroll
  for (int j = 0; j < 8; ++j) {
    int i = tid + 512*j;                       // 0..4095
    int u = i >> 3, cp = i & 7;                // row, column-pair
    float4 x = base4[(size_t)u * 256 + cp];
    lds[(2*cp  )*CPITCH + u] = make_float2(x.x, x.y);
    lds[(2*cp+1)*CPITCH + u] = make_float2(x.z, x.w);
  }
  __syncthreads();
  fft512_wave<DIR>(lds + wave * CPITCH, SC512);
  __syncthreads();
#pragma unroll
  for (int j = 0; j < 8; ++j) {
    int i = tid + 512*j;
    int u = i >> 3, cp = i & 7;
    float2 a = lds[(2*cp  )*CPITCH + u];
    float2 b = lds[(2*cp+1)*CPITCH + u];
    base4[(size_t)u * 256 + cp] = make_float4(a.x, a.y, b.x, b.y);
  }
}

// ---- inverse FFT of rows, emit real part -----------------------------------
__global__ __launch_bounds__(128) void k_fft_rows_inv(const float2* __restrict__ F,
                                                      float* __restrict__ out) {
  __shared__ float2 lds[WPB * 512];
  const int wave = threadIdx.x >> 5, lane = threadIdx.x & 31;
  const size_t row = (size_t)blockIdx.x * WPB + wave;
  float2* buf = lds + wave * 512;
  const float4* src4 = (const float4*)(F + row * 512);
#pragma unroll
  for (int j = 0; j < 8; ++j) {
    float4 x = src4[lane + 32*j];
    buf[2*(lane + 32*j)    ] = make_float2(x.x, x.y);
    buf[2*(lane + 32*j) + 1] = make_float2(x.z, x.w);
  }
  fft512_wave<1>(buf, SC512);
  float4* dst4 = (float4*)(out + row * 512);    // 4 reals per float4
#pragma unroll
  for (int j = 0; j < 4; ++j) {
    int e = 4*(lane + 32*j);
    dst4[lane + 32*j] = make_float4(buf[e].x, buf[e+1].x, buf[e+2].x, buf[e+3].x);
  }
}

// ---- radius from UNSHIFTED index (fftshift folded into index map) ----------
__device__ __forceinline__ int radius_of(int u, int v) {
  float dy = (float)((u < 256) ? u : u - 512);
  float dx = (float)((v < 256) ? v : v - 512);
  return (int)sqrtf(dy*dy + dx*dx);
}

// ---- per-(b,c) radial amplitude sums (257 bins, bin 256 = dump) ------------
__global__ __launch_bounds__(256) void k_radial(const float2* __restrict__ F,
                                                float* __restrict__ gsums) {
  __shared__ float bins[257];
  const int bc = blockIdx.x, chunk = blockIdx.y, tid = threadIdx.x;
  for (int i = tid; i < 257; i += 256) bins[i] = 0.0f;
  __syncthreads();
  const float2* Fb = F + (size_t)bc * 262144;
  const int p0 = chunk * 4096;
#pragma unroll
  for (int j = 0; j < 16; ++j) {
    int p = p0 + tid + 256*j;
    int r = radius_of(p >> 9, p & 511);
    float2 f = Fb[p];
    atomicAdd(&bins[min(r, 256)], sqrtf(f.x*f.x + f.y*f.y));
  }
  __syncthreads();
  for (int i = tid; i < 257; i += 256) atomicAdd(&gsums[bc*257 + i], bins[i]);
}

// ---- radius-bin populations (shared by all images) -------------------------
__global__ __launch_bounds__(256) void k_counts(float* __restrict__ counts) {
  __shared__ float bins[257];
  const int tid = threadIdx.x;
  for (int i = tid; i < 257; i += 256) bins[i] = 0.0f;
  __syncthreads();
  for (int p = tid; p < 262144; p += 256) {
    int r = radius_of(p >> 9, p & 511);
    atomicAdd(&bins[min(r, 256)], 1.0f);
  }
  __syncthreads();
  for (int i = tid; i < 257; i += 256) counts[i] = bins[i];
}

// ---- perturbed profile: P_pert = ((1-g)*Pu/(E+eps) + g*prior/(S+eps)) * E --
__global__ __launch_bounds__(256) void k_profile(const float* __restrict__ gsums,
                                                 const float* __restrict__ counts,
                                                 const float* __restrict__ prior,
                                                 float* __restrict__ pp) {
  __shared__ float red[256];
  const int bc = blockIdx.x, i = threadIdx.x;
  float pu = gsums[bc*257 + i] / fmaxf(counts[i], 1.0f);
  red[i] = pu; __syncthreads();
  for (int s = 128; s > 0; s >>= 1) { if (i < s) red[i] += red[i+s]; __syncthreads(); }
  float E = red[0]; __syncthreads();
  float pr = prior[i];
  red[i] = pr; __syncthreads();
  for (int s = 128; s > 0; s >>= 1) { if (i < s) red[i] += red[i+s]; __syncthreads(); }
  float S = red[0];
  float pn = (1.0f - GAMMA_C) * (pu / (E + EPS_C)) + GAMMA_C * (pr / (S + EPS_C));
  pp[bc*256 + i] = pn * E;
}

// ---- replace amplitude, keep phase:  F *= P_pert(r)/|F|  -------------------
__global__ __launch_bounds__(256) void k_scale(float2* __restrict__ F,
                                               const float* __restrict__ pp) {
  const size_t idx = (size_t)blockIdx.x * 256 + threadIdx.x;   // < 96*2^18
  const int bc = (int)(idx >> 18);
  const int p = (int)(idx & 262143);
  int r = radius_of(p >> 9, p & 511);
  float tgt = pp[bc*256 + min(r, 255)];
  float2 f = F[idx];
  float amp = sqrtf(f.x*f.x + f.y*f.y);
  if (amp > 0.0f) {
    float s = tgt / amp;
    F[idx] = make_float2(f.x*s, f.y*s);
  } else {
    F[idx] = make_float2(tgt, 0.0f);   // angle(0)=0 -> e^{i0}=1
  }
}

__global__ void k_zero(float* __restrict__ p, int n) {
  int i = blockIdx.x * 256 + threadIdx.x;
  if (i < n) p[i] = 0.0f;
}

// ---------------------------------------------------------------------------
extern "C" void kernel_launch(void* const* d_in, const int* in_sizes, int n_in,
                              void* d_out, int out_size, void* d_ws, size_t ws_size,
                              hipStream_t stream) {
  (void)in_sizes; (void)n_in; (void)out_size; (void)ws_size;
  const float* img   = (const float*)d_in[0];   // [32,3,512,512] f32
  const float* prior = (const float*)d_in[1];   // [256] f32
  float* out = (float*)d_out;

  char* ws = (char*)d_ws;
  float2* F = (float2*)ws;                               // 96*512*512 c64 = 192 MB
  const size_t offF = (size_t)96 * 262144 * sizeof(float2);
  float* gsums  = (float*)(ws + offF);                   // 96*257
  float* counts = gsums + 96*257;                        // 257 (+pad)
  float* pp     = counts + 260;                          // 96*256

  const int rowBlocks = (96 * 512) / WPB;                // 12288
  const int colBlocks = 96 * (512 / CPB);                // 3072

  k_zero<<<(96*257 + 255)/256, 256, 0, stream>>>(gsums, 96*257);
  k_fft_rows_fwd<<<rowBlocks, 128, 0, stream>>>(img, F);
  k_fft_cols<-1><<<colBlocks, 512, 0, stream>>>(F);
  k_radial<<<dim3(96, 64), 256, 0, stream>>>(F, gsums);
  k_counts<<<1, 256, 0, stream>>>(counts);
  k_profile<<<96, 256, 0, stream>>>(gsums, counts, prior, pp);
  k_scale<<<98304, 256, 0, stream>>>(F, pp);
  k_fft_cols<1><<<colBlocks, 512, 0, stream>>>(F);
  k_fft_rows_inv<<<rowBlocks, 128, 0, stream>>>(F, out);
}